// FullGN_55688545960167
// MI455X (gfx1250) — compile-verified
//
#include <hip/hip_runtime.h>
#include <hip/hip_bf16.h>

typedef __attribute__((ext_vector_type(2))) float v2f;
typedef __attribute__((ext_vector_type(8))) float v8f;

#define DD      64
#define NNODES  100000
#define NEDGES  1250000
#define NT_NODE (NNODES / 16)   // 6250 16-row tiles
#define NT_EDGE (NEDGES / 16)   // 78125 16-row tiles

// ---------------------------------------------------------------------------
// Zero-init the aggregation buffers (scatter-max identity = 0.0f, matching the
// reference's "empty segment -> 0" semantics).
// ---------------------------------------------------------------------------
__global__ void zero_kernel(float* __restrict__ p, long n) {
    long i = (long)blockIdx.x * blockDim.x + threadIdx.x;
    if (i < n) p[i] = 0.0f;
}

// ---------------------------------------------------------------------------
// Wave-level 16x64 GEMM tile via V_WMMA_F32_16X16X4_F32.
// A 16x4 f32 layout (ISA 7.12.2): lanes 0-15 hold M=0..15 / K={k0,k0+1},
// lanes 16-31 hold M=0..15 / K={k0+2,k0+3}.  C/D: VGPR v <-> row v+8*(lane>>4),
// col = n0 + (lane&15).
//
// Weights are staged in LDS *pair-interleaved*:
//     lds[(k>>1)*128 + 2*n + (k&1)] = W[k][n]
// so the (K=kr, K=kr+1) pair a lane needs for its column is one contiguous,
// 8-byte-aligned v2f -> a single ds_load_b64 lands directly in the even VGPR
// pair WMMA wants (no 2addr splits, no register-glue moves).
// ---------------------------------------------------------------------------
__device__ __forceinline__ void load_a_regs(v2f a[16], const float* __restrict__ src,
                                            int r0, int lq, int half) {
    const float* arow = src + (size_t)(r0 + lq) * DD + 2 * half;
#pragma unroll
    for (int k = 0; k < 16; ++k) a[k] = *(const v2f*)(arow + 4 * k);
}

__device__ __forceinline__ void load_a_regs_nt(v2f a[16], const float* __restrict__ src,
                                               int r0, int lq, int half) {
    const float* arow = src + (size_t)(r0 + lq) * DD + 2 * half;
#pragma unroll
    for (int k = 0; k < 16; ++k)
        a[k] = __builtin_nontemporal_load((const v2f*)(arow + 4 * k));
}

__device__ __forceinline__ v8f gemm_tile16(const v2f a[16], const float* __restrict__ Wp,
                                           int n0, int lq, int half, v8f acc) {
    const float* base = Wp + 2 * (n0 + lq) + half * 2 * DD;  // half selects K-pair {+2,+3}
#pragma unroll
    for (int k = 0; k < 16; ++k) {
        v2f b = *(const v2f*)(base + k * 4 * DD);            // pair row 2k(+half)
        acc = __builtin_amdgcn_wmma_f32_16x16x4_f32(false, a[k], false, b,
                                                    (short)0, acc, false, false);
    }
    return acc;
}

// Stage row-major W[64][64] into pair-interleaved LDS layout.
__device__ __forceinline__ void stage_w_pairs(float* __restrict__ lds,
                                              const float* __restrict__ g,
                                              int tid, int nthr) {
    const float4* s = (const float4*)g;
    for (int i = tid; i < DD * DD / 4; i += nthr) {
        int k  = i >> 4;            // source row (K index), 16 float4 per row
        int n4 = (i & 15) * 4;      // first column of this float4
        float4 v = s[i];
        float* dst = lds + (k >> 1) * (2 * DD) + (k & 1);
        dst[(n4 + 0) * 2] = v.x;
        dst[(n4 + 1) * 2] = v.y;
        dst[(n4 + 2) * 2] = v.z;
        dst[(n4 + 3) * 2] = v.w;
    }
}

// ---------------------------------------------------------------------------
// Kernel 1: fs = node @ W_s + b_s ; fr = node @ W_r + b_r
// 128 threads = 4 waves; each wave owns one 16-row tile.
// ---------------------------------------------------------------------------
__launch_bounds__(128)
__global__ void node_proj_kernel(const float* __restrict__ node,
                                 const float* __restrict__ Ws, const float* __restrict__ bs,
                                 const float* __restrict__ Wr, const float* __restrict__ br,
                                 float* __restrict__ fs, float* __restrict__ fr) {
    __shared__ float lWs[DD * DD];
    __shared__ float lWr[DD * DD];
    int tid = threadIdx.x;
    stage_w_pairs(lWs, Ws, tid, 128);
    stage_w_pairs(lWr, Wr, tid, 128);
    __syncthreads();

    int wave = tid >> 5, lane = tid & 31, half = lane >> 4, lq = lane & 15;
    int tile = blockIdx.x * 4 + wave;
    if (tile >= NT_NODE) return;
    int r0 = tile * 16;

    v2f a[16];
    load_a_regs(a, node, r0, lq, half);

    const float* lw   = lWs;
    const float* bias = bs;
    float*       out  = fs;
#pragma unroll
    for (int p = 0; p < 2; ++p) {
#pragma unroll
        for (int nt = 0; nt < 4; ++nt) {
            int n0 = nt * 16;
            float bv = bias[n0 + lq];
            v8f acc = {bv, bv, bv, bv, bv, bv, bv, bv};
            acc = gemm_tile16(a, lw, n0, lq, half, acc);
            float* orow = out + (size_t)(r0 + 8 * half) * DD + n0 + lq;
#pragma unroll
            for (int v = 0; v < 8; ++v) orow[(size_t)v * DD] = acc[v];
        }
        lw = lWr; bias = br; out = fr;
    }
}

// ---------------------------------------------------------------------------
// Kernel 2 (fused, dominant): per 16-edge tile compute
//   e = relu(edge @ W_e + b_e + fs[snd] + fr[rcv])
// and scatter-max into agg_out[snd], agg_in[rcv] via uint atomicMax on the
// bit pattern (exact for non-negative floats).  edge_features (320 MB) is
// streamed once with non-temporal loads; fs/fr/agg (102 MB) live in L2.
// ---------------------------------------------------------------------------
__launch_bounds__(128)
__global__ void edge_kernel(const float* __restrict__ edgef,
                            const int* __restrict__ senders,
                            const int* __restrict__ receivers,
                            const float* __restrict__ We, const float* __restrict__ be,
                            const float* __restrict__ fs, const float* __restrict__ fr,
                            unsigned int* __restrict__ agg_in,
                            unsigned int* __restrict__ agg_out) {
    __shared__ float lW[DD * DD];
    int tid = threadIdx.x;
    stage_w_pairs(lW, We, tid, 128);
    __syncthreads();

    int wave = tid >> 5, lane = tid & 31, half = lane >> 4, lq = lane & 15;
    int tile = blockIdx.x * 4 + wave;
    if (tile >= NT_EDGE) return;
    int e0 = tile * 16;

    v2f a[16];
    load_a_regs_nt(a, edgef, e0, lq, half);

    int snd[8], rcv[8];
#pragma unroll
    for (int v = 0; v < 8; ++v) {
        int e = e0 + 8 * half + v;     // row owned by C/D element v on this lane
        snd[v] = senders[e];
        rcv[v] = receivers[e];
    }

#pragma unroll
    for (int nt = 0; nt < 4; ++nt) {
        int n0 = nt * 16;
        float bv = be[n0 + lq];
        v8f acc = {bv, bv, bv, bv, bv, bv, bv, bv};
        acc = gemm_tile16(a, lW, n0, lq, half, acc);
        int col = n0 + lq;
#pragma unroll
        for (int v = 0; v < 8; ++v) {
            float x = acc[v] + fs[(size_t)snd[v] * DD + col]
                             + fr[(size_t)rcv[v] * DD + col];
            x = fmaxf(x, 0.0f);                       // relu -> x >= +0.0
            unsigned int ux = __float_as_uint(x);     // uint order == float order for x>=0
            atomicMax(&agg_out[(size_t)snd[v] * DD + col], ux);
            atomicMax(&agg_in [(size_t)rcv[v] * DD + col], ux);
        }
    }
}

// ---------------------------------------------------------------------------
// Kernel 3: nodes = node@W_gn + agg_in@W_gin + agg_out@W_gout + (sum of biases)
// Three GEMMs share one accumulator; three weights staged in 48 KB LDS.
// ---------------------------------------------------------------------------
__launch_bounds__(128)
__global__ void node_out_kernel(const float* __restrict__ node,
                                const float* __restrict__ aggin,
                                const float* __restrict__ aggout,
                                const float* __restrict__ Wgn,  const float* __restrict__ bgn,
                                const float* __restrict__ Wgin, const float* __restrict__ bgin,
                                const float* __restrict__ Wgout,const float* __restrict__ bgout,
                                float* __restrict__ out) {
    __shared__ float lWn[DD * DD];
    __shared__ float lWi[DD * DD];
    __shared__ float lWo[DD * DD];
    int tid = threadIdx.x;
    stage_w_pairs(lWn, Wgn, tid, 128);
    stage_w_pairs(lWi, Wgin, tid, 128);
    stage_w_pairs(lWo, Wgout, tid, 128);
    __syncthreads();

    int wave = tid >> 5, lane = tid & 31, half = lane >> 4, lq = lane & 15;
    int tile = blockIdx.x * 4 + wave;
    if (tile >= NT_NODE) return;
    int r0 = tile * 16;

    v2f an[16], ai[16], ao[16];
    load_a_regs(an, node,   r0, lq, half);
    load_a_regs(ai, aggin,  r0, lq, half);
    load_a_regs(ao, aggout, r0, lq, half);

#pragma unroll
    for (int nt = 0; nt < 4; ++nt) {
        int n0 = nt * 16;
        float bv = bgn[n0 + lq] + bgin[n0 + lq] + bgout[n0 + lq];
        v8f acc = {bv, bv, bv, bv, bv, bv, bv, bv};
        acc = gemm_tile16(an, lWn, n0, lq, half, acc);
        acc = gemm_tile16(ai, lWi, n0, lq, half, acc);
        acc = gemm_tile16(ao, lWo, n0, lq, half, acc);
        float* orow = out + (size_t)(r0 + 8 * half) * DD + n0 + lq;
#pragma unroll
        for (int v = 0; v < 8; ++v) orow[(size_t)v * DD] = acc[v];
    }
}

// ---------------------------------------------------------------------------
extern "C" void kernel_launch(void* const* d_in, const int* in_sizes, int n_in,
                              void* d_out, int out_size, void* d_ws, size_t ws_size,
                              hipStream_t stream) {
    const float* edge_features = (const float*)d_in[0];
    const float* node_features = (const float*)d_in[1];
    const int*   senders       = (const int*)d_in[2];
    const int*   receivers     = (const int*)d_in[3];
    const float* W_e   = (const float*)d_in[4];
    const float* b_e   = (const float*)d_in[5];
    const float* W_s   = (const float*)d_in[6];
    const float* b_s   = (const float*)d_in[7];
    const float* W_r   = (const float*)d_in[8];
    const float* b_r   = (const float*)d_in[9];
    const float* W_gn  = (const float*)d_in[10];
    const float* b_gn  = (const float*)d_in[11];
    const float* W_gin = (const float*)d_in[12];
    const float* b_gin = (const float*)d_in[13];
    const float* W_gout= (const float*)d_in[14];
    const float* b_gout= (const float*)d_in[15];

    // Workspace layout (floats): fs | fr | agg_in | agg_out  (= 102.4 MB)
    float* fs      = (float*)d_ws;
    float* fr      = fs + (size_t)NNODES * DD;
    float* agg_in  = fr + (size_t)NNODES * DD;
    float* agg_out = agg_in + (size_t)NNODES * DD;

    long nzero = (long)2 * NNODES * DD;   // agg_in + agg_out (contiguous)
    zero_kernel<<<(int)((nzero + 255) / 256), 256, 0, stream>>>(agg_in, nzero);

    node_proj_kernel<<<(NT_NODE + 3) / 4, 128, 0, stream>>>(
        node_features, W_s, b_s, W_r, b_r, fs, fr);

    edge_kernel<<<(NT_EDGE + 3) / 4, 128, 0, stream>>>(
        edge_features, senders, receivers, W_e, b_e, fs, fr,
        (unsigned int*)agg_in, (unsigned int*)agg_out);

    node_out_kernel<<<(NT_NODE + 3) / 4, 128, 0, stream>>>(
        node_features, agg_in, agg_out,
        W_gn, b_gn, W_gin, b_gin, W_gout, b_gout, (float*)d_out);
}